// AttnDecoderRNN_84361747628161
// MI455X (gfx1250) — compile-verified
//
#include <hip/hip_runtime.h>
#include <hip/hip_bf16.h>
#include <math.h>

// ---------------- problem constants (match reference) ----------------
#define Hdim 2048
#define Edim 1024
#define Vdim 50257
#define Sdim 2048
#define Xdim (Hdim + Edim)   // 3072

// d_out layout (floats): logits | h_new | c_new | weights
#define OUT_LOGITS 0
#define OUT_H      (Vdim)                 // 50257
#define OUT_C      (Vdim + Hdim)          // 52305
#define OUT_W      (Vdim + 2 * Hdim)      // 54353

// workspace layout (floats)
#define WS_P   0                          // 8 x 2048 partial column sums
#define WS_X   (8 * Hdim)                 // 3072  (x = [embed, enc_ctx])
#define WS_G   (WS_X + Xdim)              // 4 x 2048 gate pre-activations
#define WS_HN  (WS_G + 4 * Hdim)          // 2048  (h_new copy for logits)

typedef float v2f __attribute__((ext_vector_type(2)));
typedef float v8f __attribute__((ext_vector_type(8)));

// ---------------------------------------------------------------------
// Stage A: partial column sums of encoder_outputs.
// grid (8 col-blocks, 8 row-chunks) x 256 threads; fully coalesced b32 loads.
__global__ void enc_colsum_partial(const float* __restrict__ enc,
                                   float* __restrict__ ws) {
    int col   = blockIdx.x * blockDim.x + threadIdx.x;   // 0..2047
    int chunk = blockIdx.y;                              // 0..7
    int r0 = chunk * (Sdim / 8);
    float s = 0.f;
    #pragma unroll 4
    for (int r = 0; r < Sdim / 8; ++r)
        s += enc[(size_t)(r0 + r) * Hdim + col];
    ws[WS_P + chunk * Hdim + col] = s;
}

// Stage B: finish enc_ctx, build x = [emb_row, enc_ctx], weights = 1.0
__global__ void prep(const float* __restrict__ ws_in,
                     float* __restrict__ ws,
                     const float* __restrict__ emb,
                     const int* __restrict__ input_idx,
                     float* __restrict__ out) {
    int col = blockIdx.x * blockDim.x + threadIdx.x;     // 0..2047
    float s = 0.f;
    #pragma unroll
    for (int c = 0; c < 8; ++c) s += ws_in[WS_P + c * Hdim + col];
    ws[WS_X + Edim + col] = s;                           // enc_ctx part of x
    out[OUT_W + col] = 1.0f;                             // softmax over size-1 axis
    if (col < Edim) {
        int idx = input_idx[0];
        ws[WS_X + col] = emb[(size_t)idx * Edim + col];  // embedding part of x
    }
}

// ---------------------------------------------------------------------
// Dual-K WMMA GEMV: out[r] = dot(W1[r,:],v1) + b1[r] (+ dot(W2[r,:],v2) + b2[r])
// One wave computes 16 output rows via chained V_WMMA_F32_16X16X4_F32.
//   A (16x4, M-broadcast of the vector):  lanes 0-15 -> K=0,1 ; lanes 16-31 -> K=2,3
//   B (4x16, weight rows as columns):     lane n / lane 16+n hold W[row+n] at the
//                                         same per-lane K offsets as A, so the
//                                         contraction is layout-consistent.
// D VGPR0 then holds the result for row row_base+(lane&15) in every lane.
__global__ void wmma_gemv2(const float* __restrict__ W1, int K1,
                           const float* __restrict__ v1g,
                           const float* __restrict__ W2, int K2,
                           const float* __restrict__ v2g,
                           const float* __restrict__ b1,
                           const float* __restrict__ b2,
                           float* __restrict__ out, int rows) {
    extern __shared__ float smem[];                      // K1 (+K2) floats
    const int tid = threadIdx.x;
    for (int i = tid; i < K1; i += blockDim.x) smem[i] = v1g[i];
    if (K2)
        for (int i = tid; i < K2; i += blockDim.x) smem[K1 + i] = v2g[i];
    __syncthreads();

    const int lane = tid & 31;
    const int wave = tid >> 5;                           // 8 waves / block
    const int tile = blockIdx.x * (blockDim.x >> 5) + wave;
    const int row_base = tile * 16;
    const int n   = lane & 15;
    const int off = (lane & 16) ? 2 : 0;                 // K sub-offset per half-wave
    int r = row_base + n; if (r > rows - 1) r = rows - 1; // clamp tail (loads only)

    v8f acc = {};
    {
        const float* wrow = W1 + (size_t)r * K1;
        for (int k = 0; k < K1; k += 4) {
            float2 wv = *(const float2*)(wrow + k + off);    // global_load_b64
            float2 av = *(const float2*)(&smem[k + off]);    // ds_load_b64
            v2f A; A.x = av.x; A.y = av.y;
            v2f B; B.x = wv.x; B.y = wv.y;
            acc = __builtin_amdgcn_wmma_f32_16x16x4_f32(
                false, A, false, B, (short)0, acc, false, false);
        }
    }
    if (K2) {
        const float* wrow = W2 + (size_t)r * K2;
        const float* vsh  = smem + K1;
        for (int k = 0; k < K2; k += 4) {
            float2 wv = *(const float2*)(wrow + k + off);
            float2 av = *(const float2*)(&vsh[k + off]);
            v2f A; A.x = av.x; A.y = av.y;
            v2f B; B.x = wv.x; B.y = wv.y;
            acc = __builtin_amdgcn_wmma_f32_16x16x4_f32(
                false, A, false, B, (short)0, acc, false, false);
        }
    }

    const int orow = row_base + n;
    if (lane < 16 && orow < rows) {
        float res = acc[0] + b1[orow];
        if (b2) res += b2[orow];
        out[orow] = res;
    }
}

// ---------------------------------------------------------------------
// LSTM cell combine: c = sig(i)*tanh(g) + c_prev*sig(f); h = sig(o)*tanh(c)
__device__ __forceinline__ float sigf(float x) { return 1.0f / (1.0f + __expf(-x)); }

__global__ void lstm_combine(const float* __restrict__ ws,
                             const float* __restrict__ c_prev,
                             float* __restrict__ out,
                             float* __restrict__ ws_mut) {
    int i = blockIdx.x * blockDim.x + threadIdx.x;       // 0..2047
    float gf = ws[WS_G + 0 * Hdim + i];
    float gi = ws[WS_G + 1 * Hdim + i];
    float gg = ws[WS_G + 2 * Hdim + i];
    float go = ws[WS_G + 3 * Hdim + i];
    float c  = sigf(gi) * tanhf(gg) + c_prev[i] * sigf(gf);
    float h  = sigf(go) * tanhf(c);
    out[OUT_C + i] = c;
    out[OUT_H + i] = h;
    ws_mut[WS_HN + i] = h;
}

// ---------------------------------------------------------------------
extern "C" void kernel_launch(void* const* d_in, const int* in_sizes, int n_in,
                              void* d_out, int out_size, void* d_ws, size_t ws_size,
                              hipStream_t stream) {
    const int*   input_idx = (const int*)  d_in[0];
    const float* hidden    = (const float*)d_in[1];
    const float* context   = (const float*)d_in[2];
    const float* enc       = (const float*)d_in[3];
    const float* emb       = (const float*)d_in[4];
    const float* Wf_x = (const float*)d_in[5];  const float* bf_x = (const float*)d_in[6];
    const float* Wf_h = (const float*)d_in[7];  const float* bf_h = (const float*)d_in[8];
    const float* Wi_x = (const float*)d_in[9];  const float* bi_x = (const float*)d_in[10];
    const float* Wi_h = (const float*)d_in[11]; const float* bi_h = (const float*)d_in[12];
    const float* Wg_x = (const float*)d_in[13]; const float* bg_x = (const float*)d_in[14];
    const float* Wg_h = (const float*)d_in[15]; const float* bg_h = (const float*)d_in[16];
    const float* Wo_x = (const float*)d_in[17]; const float* bo_x = (const float*)d_in[18];
    const float* Wo_h = (const float*)d_in[19]; const float* bo_h = (const float*)d_in[20];
    // d_in[21]=Wattn, d_in[22]=battn: dead code (softmax over size-1 axis == 1)
    const float* Wout = (const float*)d_in[23];
    const float* bout = (const float*)d_in[24];

    float* out = (float*)d_out;
    float* ws  = (float*)d_ws;

    // 1) enc_ctx = column sums of encoder_outputs (16.8 MB stream)
    enc_colsum_partial<<<dim3(Hdim / 256, 8), 256, 0, stream>>>(enc, ws);
    // 2) finish sum, build x = [embed, enc_ctx], write weights = 1.0
    prep<<<Hdim / 256, 256, 0, stream>>>(ws, ws, emb, input_idx, out);

    // 3) four gate GEMVs (pre-activation), 16 rows/wave, 8 waves/block
    const float* Wx[4] = {Wf_x, Wi_x, Wg_x, Wo_x};
    const float* Wh[4] = {Wf_h, Wi_h, Wg_h, Wo_h};
    const float* bx[4] = {bf_x, bi_x, bg_x, bo_x};
    const float* bh[4] = {bf_h, bi_h, bg_h, bo_h};
    const int gateBlocks = (Hdim / 16) / 8;              // 16
    const size_t gateLds = (size_t)(Xdim + Hdim) * sizeof(float);  // 20 KB
    for (int g = 0; g < 4; ++g) {
        wmma_gemv2<<<gateBlocks, 256, gateLds, stream>>>(
            Wx[g], Xdim, ws + WS_X,
            Wh[g], Hdim, hidden,
            bx[g], bh[g],
            ws + WS_G + g * Hdim, Hdim);
    }

    // 4) LSTM cell update -> h_new, c_new
    lstm_combine<<<Hdim / 256, 256, 0, stream>>>(ws, context, out, ws);

    // 5) logits = Wout @ h_new + bout  (412 MB stream, dominates runtime)
    const int vTiles  = (Vdim + 15) / 16;                // 3142
    const int vBlocks = (vTiles + 7) / 8;                // 393
    const size_t logLds = (size_t)Hdim * sizeof(float);  // 8 KB
    wmma_gemv2<<<vBlocks, 256, logLds, stream>>>(
        Wout, Hdim, ws + WS_HN,
        nullptr, 0, nullptr,
        bout, nullptr,
        out + OUT_LOGITS, Vdim);
}